// FastNMSModule_16990890623223
// MI455X (gfx1250) — compile-verified
//
#include <hip/hip_runtime.h>

#define NCLS 80
#define TOPK 1000
#define NBINS 4096
#define CANDCAP 4096
#define SCORE_T 0.05f
#define IOU_T 0.5f

typedef float v2f __attribute__((ext_vector_type(2)));
typedef float v8f __attribute__((ext_vector_type(8)));
typedef unsigned u32x4 __attribute__((ext_vector_type(4)));
typedef int i32x4 __attribute__((ext_vector_type(4)));
typedef int i32x8 __attribute__((ext_vector_type(8)));

// ---------------------------------------------------------------------------
// Phase 1: stream scores [N,80] tiles into LDS with the Tensor Data Mover,
// then per-box threshold + max/argmax. One TDM descriptor per 128-box tile.
// TDM pad feature: pad_interval=16 DWORDs, pad_amount=1 DWORD -> effective
// LDS row stride 85 DWORDs (odd => conflict-free column reads).
// element e of the tile stream lands at LDS dword e + e/16.
// ---------------------------------------------------------------------------
__global__ void k_reduce(const float* __restrict__ scores, int N,
                         float* __restrict__ omax, int* __restrict__ ocls) {
  constexpr int TILE = 128, RS = 85;  // 80 data + 5 pad dwords per row
  __shared__ float sm[TILE * RS];     // 43520 B
  const int tid = threadIdx.x;
  const long long b0 = (long long)blockIdx.x * TILE;

#if __has_builtin(__builtin_amdgcn_tensor_load_to_lds)
  if (tid < 32) {  // wave 0 issues the DMA (EXEC ignored by tensor ops)
    int rows = N - (int)b0;
    if (rows > TILE) rows = TILE;  // OOB rows (y >= rows) read back as zero
    unsigned long long ga = (unsigned long long)(size_t)(scores + b0 * NCLS);
    u32x4 g0;
    g0[0] = 1u;                              // count=1, user mode, no gather
    g0[1] = (unsigned)(size_t)(&sm[0]);      // lds_addr
    g0[2] = (unsigned)(ga & 0xffffffffull);  // global_addr[31:0]
    g0[3] = (unsigned)((ga >> 32) & 0x01ffffffull) | (2u << 30);  // type=2
    i32x8 g1;
    // data_size=4B(2), pad_enable, pad_interval=16DW(3), pad_amount=1DW(0)
    g1[0] = (int)((2u << 16) | (1u << 20) | (3u << 22));
    g1[1] = (int)(80u << 16);                         // tensor_dim0 = 80 (lo16)
    g1[2] = (int)(((unsigned)rows & 0xffffu) << 16);  // dim0 hi | dim1 lo16
    g1[3] = (int)((((unsigned)rows >> 16) & 0xffffu) | (80u << 16)); // |tile_d0
    g1[4] = (int)TILE;                                // tile_dim1=128, dim2=0
    g1[5] = 80;                                       // tensor_dim0_stride
    g1[6] = 0;
    g1[7] = 0;
    i32x4 gz = {0, 0, 0, 0};
#if defined(__clang_major__) && (__clang_major__ >= 23)
    i32x8 gz8 = {0, 0, 0, 0, 0, 0, 0, 0};
    __builtin_amdgcn_tensor_load_to_lds(g0, g1, gz, gz, gz8, 0);
#else
    __builtin_amdgcn_tensor_load_to_lds(g0, g1, gz, gz, 0);
#endif
#if __has_builtin(__builtin_amdgcn_s_wait_tensorcnt)
    __builtin_amdgcn_s_wait_tensorcnt((short)0);
#else
    asm volatile("s_wait_tensorcnt 0x0" ::: "memory");
#endif
  }
  __syncthreads();
#else
  // Fallback: plain load/store copy loop (same padded layout)
  for (int e = tid; e < TILE * NCLS; e += blockDim.x) {
    int box = e / NCLS, c = e % NCLS;
    if (b0 + box < N) sm[box * RS + c + (c >> 4)] = scores[(b0 + box) * NCLS + c];
  }
  __syncthreads();
#endif

  long long bi = b0 + tid;
  if (bi < N) {
    float best = -1.0f; int bc = 0;
#pragma unroll 4
    for (int c = 0; c < NCLS; ++c) {
      float v = sm[tid * RS + c + (c >> 4)];
      v = (v < SCORE_T) ? 0.0f : v;        // threshold like reference
      if (v > best) { best = v; bc = c; }  // first-max (argmax semantics)
    }
    omax[bi] = best;
    ocls[bi] = bc;
  }
}

// ---------------------------------------------------------------------------
// Top-k machinery: zero, histogram, cutoff scan, compaction
// ---------------------------------------------------------------------------
__global__ void k_zero(unsigned* __restrict__ hist, unsigned* __restrict__ ctr) {
  int t = blockIdx.x * blockDim.x + threadIdx.x;
  for (int i = t; i < NBINS; i += blockDim.x * gridDim.x) hist[i] = 0;
  if (t < 8) ctr[t] = 0;
}

__global__ void k_hist(const float* __restrict__ omax, int N,
                       unsigned* __restrict__ hist) {
  int i = blockIdx.x * blockDim.x + threadIdx.x;
  if (i >= N) return;
  int bin = (int)(omax[i] * (float)NBINS);
  bin = bin < 0 ? 0 : (bin > NBINS - 1 ? NBINS - 1 : bin);
  atomicAdd(&hist[bin], 1u);
}

__global__ void k_cut(const unsigned* __restrict__ hist, unsigned* __restrict__ ctr) {
  if (threadIdx.x == 0 && blockIdx.x == 0) {
    unsigned acc = 0; int c = 0;
    for (int b = NBINS - 1; b >= 0; --b) {
      acc += hist[b];
      if (acc >= TOPK) { c = b; break; }
    }
    ctr[1] = (unsigned)c;
  }
}

__global__ void k_compact(const float* __restrict__ omax, int N,
                          unsigned* __restrict__ ctr,
                          float* __restrict__ cs, int* __restrict__ ci) {
  int i = blockIdx.x * blockDim.x + threadIdx.x;
  if (i >= N) return;
  float s = omax[i];
  int bin = (int)(s * (float)NBINS);
  bin = bin < 0 ? 0 : (bin > NBINS - 1 ? NBINS - 1 : bin);
  if ((unsigned)bin >= ctr[1]) {
    unsigned pos = atomicAdd(&ctr[0], 1u);
    if (pos < CANDCAP) { cs[pos] = s; ci[pos] = i; }
  }
}

// ---------------------------------------------------------------------------
// Bitonic sort of <=4096 candidates by (score desc, idx asc); emit top 1000.
// Deterministic: atomic append order is canonicalized by the total-order sort.
// ---------------------------------------------------------------------------
__global__ void k_sort(const float* __restrict__ cs, const int* __restrict__ ci,
                       const unsigned* __restrict__ ctr,
                       const int* __restrict__ wcls,
                       const float* __restrict__ boxes, int N,
                       int* __restrict__ sel_cls, float* __restrict__ selboxes,
                       float* __restrict__ out) {
  __shared__ float ss[CANDCAP];
  __shared__ int   si[CANDCAP];
  const int tid = threadIdx.x;
  int count = (ctr[0] < (unsigned)CANDCAP) ? (int)ctr[0] : CANDCAP;
  for (int i = tid; i < CANDCAP; i += 1024) {
    if (i < count) { ss[i] = cs[i]; si[i] = ci[i]; }
    else           { ss[i] = -1e30f; si[i] = 0x7fffffff; }
  }
  __syncthreads();
  for (int k = 2; k <= CANDCAP; k <<= 1) {
    for (int j = k >> 1; j > 0; j >>= 1) {
      for (int i = tid; i < CANDCAP; i += 1024) {
        int l = i ^ j;
        if (l > i) {
          float as = ss[i], bs = ss[l];
          int   ai = si[i], bi = si[l];
          bool desc = ((i & k) == 0);
          bool b_before_a = (bs > as) || (bs == as && bi < ai);
          bool a_before_b = (as > bs) || (as == bs && ai < bi);
          if (desc ? b_before_a : a_before_b) {
            ss[i] = bs; si[i] = bi; ss[l] = as; si[l] = ai;
          }
        }
      }
      __syncthreads();
    }
  }
  if (tid < TOPK) {
    float sc = ss[tid]; int id = si[tid];
    if (sc < -1e29f || id >= N) { sc = 0.0f; id = 0; }
    int c = wcls[id];
    out[tid]            = (float)id;   // idx
    out[TOPK + tid]     = sc;          // top_scores
    out[2 * TOPK + tid] = (float)c;    // classes
    sel_cls[tid] = c;
    selboxes[tid * 4 + 0] = boxes[(size_t)id * 4 + 0];
    selboxes[tid * 4 + 1] = boxes[(size_t)id * 4 + 1];
    selboxes[tid * 4 + 2] = boxes[(size_t)id * 4 + 2];
    selboxes[tid * 4 + 3] = boxes[(size_t)id * 4 + 3];
  }
}

// ---------------------------------------------------------------------------
// IoU matrix 1000x1000 in 16x16 wave tiles.
// Outer area-sum (area_i + area_j) on the matrix pipe via
// V_WMMA_F32_16X16X4_F32; intersection min/max on VALU; row/col broadcast via
// ds_bpermute (__shfl). EXEC stays all-ones up to the WMMA.
// ---------------------------------------------------------------------------
__global__ void k_iou(const float* __restrict__ sb, float* __restrict__ iou) {
  constexpr int TPR = 63;                           // ceil(1000/16)
  const int wave = threadIdx.x >> 5;
  const int lane = threadIdx.x & 31;
  int tile = blockIdx.x * 4 + wave;
  if (tile > TPR * TPR - 1) tile = TPR * TPR - 1;   // uniform clamp
  const int m0 = (tile / TPR) * 16;
  const int n0 = (tile % TPR) * 16;
  const int lm = lane & 15;
  int mrow = m0 + lm; if (mrow > TOPK - 1) mrow = TOPK - 1;
  int ncol = n0 + lm; if (ncol > TOPK - 1) ncol = TOPK - 1;
  float rx1 = sb[mrow * 4 + 0], ry1 = sb[mrow * 4 + 1];
  float rx2 = sb[mrow * 4 + 2], ry2 = sb[mrow * 4 + 3];
  float cx1 = sb[ncol * 4 + 0], cy1 = sb[ncol * 4 + 1];
  float cx2 = sb[ncol * 4 + 2], cy2 = sb[ncol * 4 + 3];
  const float ar = (rx2 - rx1) * (ry2 - ry1);
  const float ac = (cx2 - cx1) * (cy2 - cy1);
  const bool lo = lane < 16;
  // A (16x4): col0 = area_row, col1 = 1;  B (4x16): row0 = 1, row1 = area_col
  v2f a; a.x = lo ? ar : 0.0f; a.y = lo ? 1.0f : 0.0f;
  v2f b; b.x = lo ? 1.0f : 0.0f; b.y = lo ? ac : 0.0f;
  v8f c = {0.f, 0.f, 0.f, 0.f, 0.f, 0.f, 0.f, 0.f};
  c = __builtin_amdgcn_wmma_f32_16x16x4_f32(false, a, false, b, (short)0, c,
                                            false, false);
  cx1 = __shfl(cx1, lm, 32); cy1 = __shfl(cy1, lm, 32);
  cx2 = __shfl(cx2, lm, 32); cy2 = __shfl(cy2, lm, 32);
  const int n = n0 + lm;
  const int mbase = m0 + (lo ? 0 : 8);
#pragma unroll
  for (int r = 0; r < 8; ++r) {
    const int src = lo ? r : r + 8;        // lanes 0..15 hold row boxes
    float x1 = __shfl(rx1, src, 32), y1 = __shfl(ry1, src, 32);
    float x2 = __shfl(rx2, src, 32), y2 = __shfl(ry2, src, 32);
    float w = fminf(x2, cx2) - fmaxf(x1, cx1); w = fmaxf(w, 0.0f);
    float h = fminf(y2, cy2) - fmaxf(y1, cy1); h = fmaxf(h, 0.0f);
    float inter = w * h;
    float u = c[r] - inter + 1e-6f;        // area_i + area_j from WMMA
    int m = mbase + r;
    if (m < TOPK && n < TOPK) iou[m * TOPK + n] = inter / u;
  }
}

// ---------------------------------------------------------------------------
// Sequential suppression identical to the reference fori_loop. The next IoU
// row is prefetched into an LDS double-buffer with global_load_async_to_lds,
// overlapped via s_wait_asynccnt <= 1 (async loads complete in order).
// ---------------------------------------------------------------------------
__global__ void k_nms(const float* __restrict__ iou,
                      const int* __restrict__ sel_cls,
                      float* __restrict__ keep_out) {
  __shared__ int scls[TOPK];
  __shared__ int skeep[TOPK];
  __shared__ float srow[2][1024];
  const int j = threadIdx.x;
  if (j < TOPK) {
    scls[j] = sel_cls[j];
    skeep[j] = 1;
    // prefetch row 0
    unsigned ldsOff = (unsigned)(size_t)(&srow[0][j]);
    const float* gp = iou + j;
    asm volatile("global_load_async_to_lds_b32 %0, %1, off"
                 :: "v"(ldsOff), "v"((unsigned long long)(size_t)gp)
                 : "memory");
  }
  __syncthreads();
  for (int i = 0; i < TOPK; ++i) {
    if (i + 1 < TOPK) {  // issue next row, then wait only for the older one
      if (j < TOPK) {
        unsigned ldsOff = (unsigned)(size_t)(&srow[(i + 1) & 1][j]);
        const float* gp = iou + (size_t)(i + 1) * TOPK + j;
        asm volatile("global_load_async_to_lds_b32 %0, %1, off"
                     :: "v"(ldsOff), "v"((unsigned long long)(size_t)gp)
                     : "memory");
      }
      asm volatile("s_wait_asynccnt 0x1" ::: "memory");
    } else {
      asm volatile("s_wait_asynccnt 0x0" ::: "memory");
    }
    int ki = skeep[i];
    int ci = scls[i];
    __syncthreads();                           // all reads before writes
    if (ki && j < TOPK && scls[j] == ci)
      skeep[j] = (srow[i & 1][j] <= IOU_T) ? 1 : 0;
    __syncthreads();                           // writes before next read
  }
  if (j < TOPK) keep_out[j] = skeep[j] ? 1.0f : 0.0f;
}

// ---------------------------------------------------------------------------
extern "C" void kernel_launch(void* const* d_in, const int* in_sizes, int n_in,
                              void* d_out, int out_size, void* d_ws,
                              size_t ws_size, hipStream_t stream) {
  (void)n_in; (void)out_size; (void)ws_size;
  const float* boxes  = (const float*)d_in[0];
  const float* scores = (const float*)d_in[1];
  const int N = in_sizes[1] / NCLS;

  char* p = (char*)d_ws;
  auto take = [&](size_t bytes) -> char* {
    char* r = p;
    p += (bytes + 255) & ~(size_t)255;
    return r;
  };
  float*    ws_max   = (float*)take((size_t)N * 4);
  int*      ws_cls   = (int*)take((size_t)N * 4);
  unsigned* hist     = (unsigned*)take(NBINS * 4);
  unsigned* ctr      = (unsigned*)take(256);
  float*    cand_s   = (float*)take(CANDCAP * 4);
  int*      cand_i   = (int*)take(CANDCAP * 4);
  int*      sel_cls  = (int*)take(1024 * 4);
  float*    selboxes = (float*)take(4096 * 4);
  float*    iou      = (float*)take((size_t)TOPK * TOPK * 4);
  float*    out      = (float*)d_out;

  k_zero   <<<16, 256, 0, stream>>>(hist, ctr);
  k_reduce <<<(N + 127) / 128, 128, 0, stream>>>(scores, N, ws_max, ws_cls);
  k_hist   <<<(N + 255) / 256, 256, 0, stream>>>(ws_max, N, hist);
  k_cut    <<<1, 32, 0, stream>>>(hist, ctr);
  k_compact<<<(N + 255) / 256, 256, 0, stream>>>(ws_max, N, ctr, cand_s, cand_i);
  k_sort   <<<1, 1024, 0, stream>>>(cand_s, cand_i, ctr, ws_cls, boxes, N,
                                    sel_cls, selboxes, out);
  k_iou    <<<(63 * 63 + 3) / 4, 128, 0, stream>>>(selboxes, iou);
  k_nms    <<<1, 1024, 0, stream>>>(iou, sel_cls, out + 3 * TOPK);
}